// ISTFT_5669356834119
// MI455X (gfx1250) — compile-verified
//
#include <hip/hip_runtime.h>
#include <math.h>

// ============================================================================
// ISTFT, restructured as a single f32 WMMA GEMM on gfx1250.
//
// Key observation: reference keeps only y[:, 512:1112) (600 samples). Frame t
// touches sample s only if 0 <= s-256t < 1024, so only frames t=0..4 of 2048
// contribute. The whole ISTFT collapses to:
//     out(64 x 600) = Xp(64 x 5130) * G(5130 x 600)
// where Xp packs Re/Im of x[b, k, t] for t in [0,5), and G fuses the irfft
// basis, Hann window, overlap-add placement, 1/N and the final 1/coeff.
//
// Effective traffic: ~1.3 MB input gather + 150 KB output + 12.5 MB basis
// build -> launch-latency bound on MI455X (23.3 TB/s). Compute ~0.4 GFLOP ->
// V_WMMA_F32_16X16X4_F32 chosen for exact f32 numerics (throughput moot).
// ============================================================================

#define B_N 64
#define F_N 513
#define T_N 2048
#define WIN 1024
#define HOP 256
#define LEN 600
#define T_USED 5                    // frames t=0..4 cover samples [512,1112)
#define K_REAL (T_USED * F_N * 2)   // 5130 (re/im interleaved per (t,k))
#define KP 5136                     // K padded: multiple of 16 (1284 k-steps of 4)
#define NP 608                      // 600 padded to 38 tiles of 16
#define S0 (WIN / 2)                // 512: first retained output sample

typedef float v2f __attribute__((ext_vector_type(2)));
typedef float v8f __attribute__((ext_vector_type(8)));

// ---------------------------------------------------------------------------
// Xp[b][kk] = x[b, k, t, d] for kk = (t*513 + k)*2 + d, zero-padded to KP.
// Input x: (64, 1, 513, 2048, 2) contiguous f32.
// ---------------------------------------------------------------------------
__global__ void pack_x_kernel(const float* __restrict__ x,
                              float* __restrict__ xp) {
  int i = blockIdx.x * blockDim.x + threadIdx.x;
  if (i >= B_N * KP) return;
  int b  = i / KP;
  int kk = i % KP;
  float v = 0.0f;
  if (kk < K_REAL) {
    int t = kk / (F_N * 2);
    int r = kk % (F_N * 2);
    int k = r >> 1;
    int d = r & 1;
    v = x[(((size_t)b * F_N + k) * T_N + t) * 2 + d];
  }
  xp[(size_t)b * KP + kk] = v;
}

// ---------------------------------------------------------------------------
// Gt[l][kk]: fused basis, stored N-major so each lane's B-fragment (two
// K-adjacent values) is one b64 load in the GEMM.
//   s = l + 512, n = s - 256 t
//   d==0:  2/1 * cos(2*pi*k*n/1024)      d==1: -2/1 * sin(2*pi*k*n/1024)
//   (factor 1 at k=0 and k=512; imag parts there vanish since sin == 0)
//   * hann(n) * (1/1024) * (1/coeff=0.5)
// Angle reduced exactly with (k*n) & 1023 to keep sinf/cosf accurate.
// ---------------------------------------------------------------------------
__global__ void basis_kernel(float* __restrict__ gt) {
  int i = blockIdx.x * blockDim.x + threadIdx.x;
  if (i >= NP * KP) return;
  int l  = i / KP;
  int kk = i % KP;
  float v = 0.0f;
  if (l < LEN && kk < K_REAL) {
    int t = kk / (F_N * 2);
    int r = kk % (F_N * 2);
    int k = r >> 1;
    int d = r & 1;
    int n = (l + S0) - HOP * t;
    if (n >= 0 && n < WIN) {
      const float step = 6.283185307179586f / (float)WIN;  // 2*pi/1024
      int   m    = (k * n) & (WIN - 1);                    // exact reduction
      float ang  = step * (float)m;
      float base = (d == 0) ? cosf(ang) : -sinf(ang);
      float ak   = (k == 0 || k == F_N - 1) ? 1.0f : 2.0f; // DC/Nyquist
      float wnd  = 0.5f - 0.5f * cosf(step * (float)n);    // hann
      v = base * ak * wnd * (1.0f / (float)WIN) * 0.5f;    // 1/N and 1/coeff
    }
  }
  gt[(size_t)l * KP + kk] = v;
}

// ---------------------------------------------------------------------------
// GEMM: out(64x600) = Xp(64xKP) * Gt^T, one wave per 16x16 tile.
// Fragment layouts per ISA 7.12.2 (f32, wave32):
//   A 16x4 : 2 VGPRs; lanes 0-15 hold M=lane, K={0,1}; lanes 16-31 K={2,3}
//   B 4x16 : 2 VGPRs; lanes 0-15 hold N=lane, K={0,1}; lanes 16-31 K={2,3}
//   C 16x16: 8 VGPRs; VGPR r = row r (lanes 0-15) / row r+8 (lanes 16-31)
// EXEC is all-1s for every wave executing WMMA (grid = exactly 152 tiles).
// ---------------------------------------------------------------------------
__global__ void istft_wmma_gemm(const float* __restrict__ xp,
                                const float* __restrict__ gt,
                                float* __restrict__ out) {
  const int MT = B_N / 16;               // 4 M-tiles
  const int NT = NP / 16;                // 38 N-tiles
  const int wave = threadIdx.x >> 5;
  const int tile = blockIdx.x * (blockDim.x >> 5) + wave;
  if (tile >= MT * NT) return;           // wave-uniform: EXEC stays full

  const int mt   = tile % MT;
  const int nt   = tile / MT;
  const int lane = threadIdx.x & 31;
  const int half = lane >> 4;            // 0: K 0/1, 1: K 2/3
  const int l16  = lane & 15;

  const float* arow = xp + (size_t)(mt * 16 + l16) * KP + 2 * half;
  const float* brow = gt + (size_t)(nt * 16 + l16) * KP + 2 * half;

  v8f acc = {};
#pragma unroll 4
  for (int kb = 0; kb < KP; kb += 4) {
    v2f a = *(const v2f*)(arow + kb);    // b64 load, K-adjacent pair
    v2f b = *(const v2f*)(brow + kb);    // b64 load, K-adjacent pair
    // (neg_a, A, neg_b, B, c_mod, C, reuse_a, reuse_b)
    acc = __builtin_amdgcn_wmma_f32_16x16x4_f32(
        false, a, false, b, (short)0, acc, false, false);
  }

  const int col = nt * 16 + l16;
  if (col < LEN) {                       // divergence only after all WMMAs
#pragma unroll
    for (int r = 0; r < 8; ++r) {
      int row = mt * 16 + r + 8 * half;
      out[(size_t)row * LEN + col] = acc[r];
    }
  }
}

// ---------------------------------------------------------------------------
// Launch. Workspace layout (floats): Xp [64*5136] then Gt [608*5136],
// total ~13.8 MB (input is 537 MB, so d_ws is comfortably larger).
// All three kernels on `stream`, dependency via stream ordering.
// ---------------------------------------------------------------------------
extern "C" void kernel_launch(void* const* d_in, const int* in_sizes, int n_in,
                              void* d_out, int out_size, void* d_ws,
                              size_t ws_size, hipStream_t stream) {
  (void)in_sizes; (void)n_in; (void)out_size; (void)ws_size;
  const float* x  = (const float*)d_in[0];
  float* out      = (float*)d_out;
  float* xp       = (float*)d_ws;
  float* gt       = xp + (size_t)B_N * KP;

  {
    int total = B_N * KP;                       // 328,704
    pack_x_kernel<<<(total + 255) / 256, 256, 0, stream>>>(x, xp);
  }
  {
    int total = NP * KP;                        // 3,122,688
    basis_kernel<<<(total + 255) / 256, 256, 0, stream>>>(gt);
  }
  // 152 tiles = 19 blocks * 8 waves, one 16x16 tile per wave
  istft_wmma_gemm<<<19, 256, 0, stream>>>(xp, gt, out);
}